// OrganismLevel_83365315215911
// MI455X (gfx1250) — compile-verified
//
#include <hip/hip_runtime.h>

// Problem constants (match reference)
#define B_DIM 2048
#define N_DIM 4096
#define M_DIM 4096
#define K_BITS 8
#define HID_DIM 64
#define TAB 256

typedef __attribute__((ext_vector_type(2))) float v2f;
typedef __attribute__((ext_vector_type(8))) float v8f;

// ---------------------------------------------------------------------------
// Kernel 1: per-row int8 absmax quantize-dequantize of ram_tables (M x 256)
// ---------------------------------------------------------------------------
__global__ void k_quant_tables(const float* __restrict__ tab,
                               float* __restrict__ dq) {
    const int m = blockIdx.x;     // row
    const int t = threadIdx.x;    // 256 threads = 256 entries
    float v = tab[m * TAB + t];
    __shared__ float red[256];
    red[t] = fabsf(v);
    __syncthreads();
    for (int s = 128; s > 0; s >>= 1) {
        if (t < s) red[t] = fmaxf(red[t], red[t + s]);
        __syncthreads();
    }
    const float scale = fmaxf(red[0], 1e-6f) / 127.0f;
    float q = rintf(v / scale);                 // round-half-even like jnp.round
    q = fminf(fmaxf(q, -127.0f), 127.0f);
    dq[m * TAB + t] = q * scale;
}

// ---------------------------------------------------------------------------
// Kernel 2: quantize-dequantize halt_w (1 x M)
// ---------------------------------------------------------------------------
__global__ void k_quant_halt(const float* __restrict__ hw,
                             float* __restrict__ dq) {
    const int t = threadIdx.x;    // 256
    float amax = 0.0f;
    for (int i = t; i < M_DIM; i += 256) amax = fmaxf(amax, fabsf(hw[i]));
    __shared__ float red[256];
    red[t] = amax;
    __syncthreads();
    for (int s = 128; s > 0; s >>= 1) {
        if (t < s) red[t] = fmaxf(red[t], red[t + s]);
        __syncthreads();
    }
    const float scale = fmaxf(red[0], 1e-6f) / 127.0f;
    for (int i = t; i < M_DIM; i += 256) {
        float q = rintf(hw[i] / scale);
        q = fminf(fmaxf(q, -127.0f), 127.0f);
        dq[i] = q * scale;
    }
}

// ---------------------------------------------------------------------------
// Kernel 3: sign bitmask of x (>0), one u32 per 32 consecutive elements.
// wave32 ballot: each wave produces one word.
// ---------------------------------------------------------------------------
__global__ void k_sign_bits(const float* __restrict__ x,
                            unsigned int* __restrict__ bits) {
    const unsigned long long gid =
        (unsigned long long)blockIdx.x * blockDim.x + threadIdx.x;
    const float v = x[gid];
    const unsigned long long m = __ballot(v > 0.0f);   // wave32: low 32 bits
    if ((threadIdx.x & 31u) == 0u) bits[gid >> 5] = (unsigned int)m;
}

// ---------------------------------------------------------------------------
// Kernel 4: RAM lookup + halt dot. One block per batch row b.
// addr[b,m] = sum_k bit(x[b, conn[m,k]]) << k ; halt[b] = sum_m tab_dq[m,addr]*hw_dq[m]
// ---------------------------------------------------------------------------
__global__ void k_ram_halt(const unsigned int* __restrict__ bits,
                           const int* __restrict__ conn,
                           const float* __restrict__ tabdq,
                           const float* __restrict__ hwdq,
                           const float* __restrict__ halt_b,
                           float* __restrict__ out) {
    const int b = blockIdx.x;
    const int t = threadIdx.x;    // 256
    __shared__ unsigned int mask[N_DIM / 32];   // 128 words = 512 B
    if (t < (N_DIM / 32)) mask[t] = bits[b * (N_DIM / 32) + t];
    __syncthreads();

    float acc = 0.0f;
    for (int m = t; m < M_DIM; m += 256) {
        const int4* cp = (const int4*)(conn + m * K_BITS);
        const int4 c0 = cp[0];
        const int4 c1 = cp[1];
        unsigned int addr = 0u;
        addr |= ((mask[((unsigned)c0.x) >> 5] >> (c0.x & 31)) & 1u) << 0;
        addr |= ((mask[((unsigned)c0.y) >> 5] >> (c0.y & 31)) & 1u) << 1;
        addr |= ((mask[((unsigned)c0.z) >> 5] >> (c0.z & 31)) & 1u) << 2;
        addr |= ((mask[((unsigned)c0.w) >> 5] >> (c0.w & 31)) & 1u) << 3;
        addr |= ((mask[((unsigned)c1.x) >> 5] >> (c1.x & 31)) & 1u) << 4;
        addr |= ((mask[((unsigned)c1.y) >> 5] >> (c1.y & 31)) & 1u) << 5;
        addr |= ((mask[((unsigned)c1.z) >> 5] >> (c1.z & 31)) & 1u) << 6;
        addr |= ((mask[((unsigned)c1.w) >> 5] >> (c1.w & 31)) & 1u) << 7;
        acc += tabdq[m * TAB + addr] * hwdq[m];
    }

    __shared__ float red[256];
    red[t] = acc;
    __syncthreads();
    for (int s = 128; s > 0; s >>= 1) {
        if (t < s) red[t] += red[t + s];
        __syncthreads();
    }
    if (t == 0) out[b * 2 + 1] = red[0] + halt_b[0];
}

// ---------------------------------------------------------------------------
// Kernel 5: engage = relu(x @ w1^T + b1) @ w2^T + b2 via V_WMMA_F32_16X16X4_F32
// Block = 128 threads (4 waves). Block owns 16 rows of x; wave w owns columns
// [16w, 16w+16) of the 64 hidden units. K loop over 4096 in steps of 4.
// A(16x4) layout: lane = 16*hi + r holds A[r][2*hi + {0,1}]
// B(4x16) layout: lane = 16*hi + r holds B[2*hi + {0,1}][r]   (B[k][n] = w1[n][k])
// C/D layout: VGPR i, lanes 0-15 -> (row i,     col lane)
//                     lanes 16-31 -> (row 8+i, col lane-16)
// ---------------------------------------------------------------------------
__global__ void __launch_bounds__(128)
k_engage_wmma(const float* __restrict__ x,
              const float* __restrict__ w1,
              const float* __restrict__ b1,
              const float* __restrict__ w2,
              const float* __restrict__ b2,
              float* __restrict__ out) {
    const int lane = threadIdx.x & 31;
    const int wave = threadIdx.x >> 5;        // 0..3 -> n-tile
    const int r    = lane & 15;
    const int hi   = lane >> 4;               // 0 or 1
    const int row0 = blockIdx.x * 16;
    const int col0 = wave * 16;

    const float* aPtr = x  + (size_t)(row0 + r) * N_DIM + 2 * hi;
    const float* bPtr = w1 + (size_t)(col0 + r) * N_DIM + 2 * hi;

    v8f acc = {};
#pragma unroll 8
    for (int k = 0; k < N_DIM; k += 4) {
        v2f a = *(const v2f*)(aPtr + k);
        v2f bb = *(const v2f*)(bPtr + k);
        // D = A(16x4,f32) x B(4x16,f32) + C(16x16,f32)
        acc = __builtin_amdgcn_wmma_f32_16x16x4_f32(
            /*neg_a=*/false, a, /*neg_b=*/false, bb,
            /*c_mod=*/(short)0, acc, /*reuse_a=*/false, /*reuse_b=*/false);
    }

    // Fused epilogue: per-lane column n = col0 + r
    const float b1v = b1[col0 + r];
    const float w2v = w2[col0 + r];
    float p[8];
#pragma unroll
    for (int i = 0; i < 8; ++i) {
        float c = acc[i] + b1v;
        c = fmaxf(c, 0.0f);                   // relu
        p[i] = c * w2v;                       // * w2[n]
    }
    // Reduce across the 16 lanes of each half (masks 1..8 stay in-half)
#pragma unroll
    for (int off = 8; off >= 1; off >>= 1) {
#pragma unroll
        for (int i = 0; i < 8; ++i) p[i] += __shfl_xor(p[i], off, 32);
    }
    // lane 0 now holds rows row0+0..7, lane 16 holds rows row0+8..15
    __shared__ float part[4][16];
    if (lane == 0) {
#pragma unroll
        for (int i = 0; i < 8; ++i) part[wave][i] = p[i];
    }
    if (lane == 16) {
#pragma unroll
        for (int i = 0; i < 8; ++i) part[wave][8 + i] = p[i];
    }
    __syncthreads();
    if (threadIdx.x < 16) {
        const float e = part[0][threadIdx.x] + part[1][threadIdx.x] +
                        part[2][threadIdx.x] + part[3][threadIdx.x] + b2[0];
        out[(row0 + threadIdx.x) * 2 + 0] = e;
    }
}

// ---------------------------------------------------------------------------
// Launcher
// ---------------------------------------------------------------------------
extern "C" void kernel_launch(void* const* d_in, const int* in_sizes, int n_in,
                              void* d_out, int out_size, void* d_ws, size_t ws_size,
                              hipStream_t stream) {
    const float* x       = (const float*)d_in[0];
    const float* tables  = (const float*)d_in[1];
    const float* w1      = (const float*)d_in[2];
    const float* b1      = (const float*)d_in[3];
    const float* w2      = (const float*)d_in[4];
    const float* b2      = (const float*)d_in[5];
    const float* halt_w  = (const float*)d_in[6];
    const float* halt_b  = (const float*)d_in[7];
    const int*   conn    = (const int*)d_in[8];
    float* out = (float*)d_out;

    // Workspace layout
    char* ws = (char*)d_ws;
    float* tabdq      = (float*)(ws);                                   // 4 MB
    float* hwdq       = (float*)(ws + (size_t)M_DIM * TAB * 4);         // 16 KB
    unsigned int* bits = (unsigned int*)(ws + (size_t)M_DIM * TAB * 4 + (size_t)M_DIM * 4); // 1 MB

    // 1) quantize-dequantize tables and halt_w
    k_quant_tables<<<M_DIM, 256, 0, stream>>>(tables, tabdq);
    k_quant_halt<<<1, 256, 0, stream>>>(halt_w, hwdq);

    // 2) sign bitmask of x
    const int total = B_DIM * N_DIM;                 // 8,388,608
    k_sign_bits<<<total / 256, 256, 0, stream>>>(x, bits);

    // 3) RAM lookup + halt dot (one block per batch row)
    k_ram_halt<<<B_DIM, 256, 0, stream>>>(bits, conn, tabdq, hwdq, halt_b, out);

    // 4) engage GEMM on WMMA f32 (128 m-tiles x 4 n-tile waves)
    k_engage_wmma<<<B_DIM / 16, 128, 0, stream>>>(x, w1, b1, w2, b2, out);
}